// GptOssTopKRouter_3375844295434
// MI455X (gfx1250) — compile-verified
//
#include <hip/hip_runtime.h>
#include <math.h>

// GPT-OSS Top-K router for MI455X (gfx1250, wave32, WMMA, async-to-LDS staging).
// logits = x @ W^T + b ; top-4 ; softmax(top4) ; scatter dense + indices.

#define HIDDEN 2880
#define NEXP   32
#define TOPK   4
#define KP     240                     // K panel staged in LDS (double buffered)
#define PANELS (HIDDEN / KP)           // 12
#define LDSTR  244                     // padded row stride (floats): 16B aligned, 64-bank clean
#define WAVES  4
#define TOKENS_PER_BLOCK (WAVES * 16)  // 64
#define F4_PER_THREAD (NEXP * (KP / 4) / (WAVES * 32))  // 1920 / 128 = 15

typedef float v2f __attribute__((ext_vector_type(2)));
typedef float v4f __attribute__((ext_vector_type(4)));
typedef float v8f __attribute__((ext_vector_type(8)));
typedef int   v4i __attribute__((ext_vector_type(4)));

__global__ __launch_bounds__(128, 4) void gptoss_topk_router_kernel(
    const float* __restrict__ x,      // [T, HIDDEN]
    const float* __restrict__ W,      // [NEXP, HIDDEN]
    const float* __restrict__ bias,   // [NEXP]
    float* __restrict__ scores,       // [T, NEXP]
    int*   __restrict__ indices)      // [T, TOPK]
{
    __shared__ float ldsW[2][NEXP * LDSTR];              // double-buffered W K-panels
    __shared__ float ldsLog[WAVES * 16 * (NEXP + 1)];    // per-wave 16x32 logits (padded)

    const int tid   = threadIdx.x;
    const int wave  = tid >> 5;
    const int lane  = tid & 31;
    const int laneM = lane & 15;   // token row (A) / expert column (B) owned by this lane
    const int half  = lane >> 4;   // K-pair selector per fp32 WMMA A/B layout
    const int kl    = 2 * half;
    const int tokenBase = blockIdx.x * TOKENS_PER_BLOCK + wave * 16;

    const float* aRow = x + (size_t)(tokenBase + laneM) * HIDDEN;

    // ---- thread-invariant staging map: 32 rows x 4 lanes/row, 15 float4 each ----
    const int srow = tid >> 2;         // 0..31 (expert row)
    const int scol = (tid & 3) * 4;    // starting float col, stride 16 floats per step
    const float*   gstage = W + (size_t)srow * HIDDEN + scol;
    const unsigned lbase0 = (unsigned)(uintptr_t)&ldsW[0][srow * LDSTR + scol];
    const unsigned lbase1 = (unsigned)(uintptr_t)&ldsW[1][srow * LDSTR + scol];

    // Async DMA of one W K-panel into LDS buffer (p & 1). 15 instructions/wave,
    // tracked by ASYNCcnt; no VGPR round-trip, no per-load wait.
    auto stage = [&](int p) {
        const float* g = gstage + p * KP;
        unsigned l = (p & 1) ? lbase1 : lbase0;
        #pragma unroll
        for (int j = 0; j < F4_PER_THREAD; ++j) {
            asm volatile("global_load_async_to_lds_b128 %0, %1, off"
                         :: "v"(l), "v"(g) : "memory");
            g += 16;   // 16 floats = 64B (4 lanes interleave 16B each)
            l += 64;
        }
    };

    v8f acc0 = {};   // experts 0..15
    v8f acc1 = {};   // experts 16..31

    stage(0);
    for (int p = 0; p < PANELS; ++p) {
        if (p + 1 < PANELS) {
            stage(p + 1);   // prefetch next panel into the other buffer
            asm volatile("s_wait_asynccnt 0xf" ::: "memory");  // panel p landed (15 newer in flight)
        } else {
            asm volatile("s_wait_asynccnt 0x0" ::: "memory");  // last panel landed
        }
        __syncthreads();    // all waves' async stores visible

        const float* aPan  = aRow + p * KP;
        const float* bRow0 = &ldsW[p & 1][laneM * LDSTR];
        const float* bRow1 = &ldsW[p & 1][(laneM + 16) * LDSTR];
        #pragma unroll 4
        for (int c = 0; c < KP / 4; ++c) {
            // A 16x4 fp32 fragment: lane holds token row laneM, K = 4c + 2*half + {0,1}
            v2f a  = *(const v2f*)(aPan + 4 * c + kl);
            // B 4x16 fragments (B[k][n] = W[n][k]) from LDS
            v2f b0 = *(const v2f*)(bRow0 + 4 * c + kl);
            v2f b1 = *(const v2f*)(bRow1 + 4 * c + kl);
            acc0 = __builtin_amdgcn_wmma_f32_16x16x4_f32(
                       false, a, false, b0, (short)0, acc0, false, false);
            acc1 = __builtin_amdgcn_wmma_f32_16x16x4_f32(
                       false, a, false, b1, (short)0, acc1, false, false);
        }
        __syncthreads();    // buffer (p&1) free for the p+2 prefetch
    }

    // ---- bias add (C/D layout: VGPR r, lane -> M = r + 8*half, N = laneM) ----
    const float b0s = bias[laneM];
    const float b1s = bias[laneM + 16];
    #pragma unroll
    for (int r = 0; r < 8; ++r) { acc0[r] += b0s; acc1[r] += b1s; }

    // ---- stage 16x32 logits to LDS so each lane can scan one token's row ----
    float* llog = &ldsLog[wave * 16 * (NEXP + 1)];
    #pragma unroll
    for (int r = 0; r < 8; ++r) {
        const int m = r + 8 * half;
        llog[m * (NEXP + 1) + laneM]      = acc0[r];
        llog[m * (NEXP + 1) + laneM + 16] = acc1[r];
    }
    // LDS is in-order per wave; fence data return before cross-lane reads.
    asm volatile("s_wait_dscnt 0" ::: "memory");

    if (lane < 16) {
        const int token = tokenBase + lane;

        float vals[NEXP];
        #pragma unroll
        for (int e = 0; e < NEXP; ++e) vals[e] = llog[lane * (NEXP + 1) + e];

        int   idx[TOPK];
        float tv[TOPK];
        #pragma unroll
        for (int k = 0; k < TOPK; ++k) {
            float best = -INFINITY; int bi = 0;
            #pragma unroll
            for (int e = 0; e < NEXP; ++e) {
                if (vals[e] > best) { best = vals[e]; bi = e; }  // first-max tie-break
            }
            tv[k] = best; idx[k] = bi;
            #pragma unroll
            for (int e = 0; e < NEXP; ++e) if (e == bi) vals[e] = -INFINITY;
        }

        // softmax over the 4 selected logits (max is tv[0])
        const float m = tv[0];
        float w[TOPK]; float sum = 0.f;
        #pragma unroll
        for (int k = 0; k < TOPK; ++k) { w[k] = __expf(tv[k] - m); sum += w[k]; }
        const float inv = 1.f / sum;
        #pragma unroll
        for (int k = 0; k < TOPK; ++k) w[k] *= inv;

        // dense scatter row
        float dense[NEXP];
        #pragma unroll
        for (int e = 0; e < NEXP; ++e) {
            float v = 0.f;
            #pragma unroll
            for (int k = 0; k < TOPK; ++k) v = (e == idx[k]) ? w[k] : v;
            dense[e] = v;
        }

        float* outRow = scores + (size_t)token * NEXP;   // 128B-aligned row
        #pragma unroll
        for (int e = 0; e < NEXP; e += 4)
            *(v4f*)(outRow + e) = *(const v4f*)(&dense[e]);

        v4i iv = { idx[0], idx[1], idx[2], idx[3] };
        *(v4i*)(indices + (size_t)token * TOPK) = iv;
    }
}

extern "C" void kernel_launch(void* const* d_in, const int* in_sizes, int n_in,
                              void* d_out, int out_size, void* d_ws, size_t ws_size,
                              hipStream_t stream) {
    const float* x    = (const float*)d_in[0];   // [4,4096,2880] fp32
    const float* W    = (const float*)d_in[1];   // [32,2880] fp32
    const float* bias = (const float*)d_in[2];   // [32] fp32

    const int T = in_sizes[0] / HIDDEN;          // 16384 tokens

    float* scores  = (float*)d_out;                            // [T,32] fp32
    int*   indices = (int*)((float*)d_out + (size_t)T * NEXP); // [T,4] int32, concatenated

    dim3 grid(T / TOKENS_PER_BLOCK);             // 16384 / 64 = 256 blocks
    gptoss_topk_router_kernel<<<grid, 128, 0, stream>>>(x, W, bias, scores, indices);
}